// spatial_attn_37864431682520
// MI455X (gfx1250) — compile-verified
//
#include <hip/hip_runtime.h>
#include <hip/hip_bf16.h>

typedef __attribute__((ext_vector_type(2))) float v2f;
typedef __attribute__((ext_vector_type(8))) float v8f;

#define NN 128
#define LSTR 132   // padded LDS row stride (floats): breaks bank alignment for column walks

__launch_bounds__(256, 1)
__global__ void gat_attn_kernel(const float* __restrict__ A,
                                const float* __restrict__ W,
                                const float* __restrict__ b,
                                float* __restrict__ out) {
  __shared__ float sM[NN * LSTR];    // M[r][c] = A[t][r][c]
  __shared__ float sAl[NN * LSTR];   // alpha[i][j]
  __shared__ float sw[2 * NN];       // w1 | w2
  __shared__ float sS[2 * NN];       // s1 | s2

  const int t   = blockIdx.x;
  const int tid = threadIdx.x;
  const float* At = A + (size_t)t * NN * NN;

  // ---- Phase 1: stage A[t] (64 KB) into LDS, float4-vectorized ----
  for (int v = tid; v < NN * NN / 4; v += 256) {
    const int r  = v / (NN / 4);
    const int c4 = (v % (NN / 4)) * 4;
    const float4 d = *(const float4*)(At + r * NN + c4);
    float* p = &sM[r * LSTR + c4];
    p[0] = d.x; p[1] = d.y; p[2] = d.z; p[3] = d.w;
  }
  if (tid < 2 * NN) sw[tid] = W[tid];
  __syncthreads();

  // ---- Phase 2: s1[i] = sum_j M[j][i]*w1[j] ; s2[i] = sum_j M[j][i]*w2[j] ----
  {
    const int i = tid & (NN - 1);
    const float* wv = &sw[(tid >> 7) * NN];   // tid<128 -> w1, tid>=128 -> w2
    float acc = 0.f;
    #pragma unroll 4
    for (int j = 0; j < NN; ++j) acc = fmaf(sM[j * LSTR + i], wv[j], acc);
    sS[tid] = acc;
  }
  __syncthreads();

  // ---- Phase 3: leaky_relu + row softmax (thread i owns row i) ----
  if (tid < NN) {
    const float si = sS[tid] + b[0];
    float m = -3.4e38f;
    for (int j = 0; j < NN; ++j) {
      float x = si + sS[NN + j];
      x = (x > 0.f) ? x : 0.2f * x;
      m = fmaxf(m, x);
    }
    float sum = 0.f;
    float* arow = &sAl[tid * LSTR];
    for (int j = 0; j < NN; ++j) {
      float x = si + sS[NN + j];
      x = (x > 0.f) ? x : 0.2f * x;
      const float ex = __expf(x - m);
      arow[j] = ex;
      sum += ex;
    }
    const float inv = 1.0f / sum;
    for (int j = 0; j < NN; ++j) arow[j] *= inv;
  }
  __syncthreads();

  // ---- Phase 4: out[i][f] = sum_j alpha[i][j] * M[f][j]  (f32 WMMA 16x16x4) ----
  // A-frag (16x4):  row = lane&15, K = vgpr + 2*(lane>=16)
  // B-frag (4x16):  col = lane&15, K = vgpr + 2*(lane>=16)   [B row k = alpha's j; B col n = f]
  // C/D  (16x16):   row = vgpr + 8*(lane>=16), col = lane&15
  const int wave  = tid >> 5;
  const int lane  = tid & 31;
  const int half  = lane >> 4;
  const int l     = lane & 15;
  const int fTile = wave * 16;          // 8 waves cover 8 f-tiles
  float* outT = out + (size_t)t * NN * NN;

  for (int q = 0; q < 8; ++q) {         // 8 i-tiles per wave
    const int iTile = q * 16;
    v8f acc = {0.f, 0.f, 0.f, 0.f, 0.f, 0.f, 0.f, 0.f};
    const float* aRow = &sAl[(iTile + l) * LSTR + 2 * half];  // alpha[iTile+l][k0+2h+{0,1}]
    const float* bRow = &sM[(fTile + l) * LSTR + 2 * half];   // M[fTile+l][k0+2h+{0,1}]
    #pragma unroll 4
    for (int k0 = 0; k0 < NN; k0 += 4) {
      v2f a;  a.x  = aRow[k0]; a.y  = aRow[k0 + 1];
      v2f bb; bb.x = bRow[k0]; bb.y = bRow[k0 + 1];
      acc = __builtin_amdgcn_wmma_f32_16x16x4_f32(
          /*neg_a=*/false, a, /*neg_b=*/false, bb,
          /*c_mod=*/(short)0, acc, /*reuse_a=*/false, /*reuse_b=*/false);
    }
    #pragma unroll
    for (int g = 0; g < 8; ++g) {
      outT[(size_t)(iTile + g + 8 * half) * NN + fTile + l] = acc[g];
    }
  }
}

extern "C" void kernel_launch(void* const* d_in, const int* in_sizes, int n_in,
                              void* d_out, int out_size, void* d_ws, size_t ws_size,
                              hipStream_t stream) {
  const float* A = (const float*)d_in[0];
  const float* W = (const float*)d_in[1];
  const float* b = (const float*)d_in[2];
  float* out = (float*)d_out;
  const int T = in_sizes[0] / (NN * NN);   // 64
  gat_attn_kernel<<<T, 256, 0, stream>>>(A, W, b, out);
}